// rrnn_37417755083229
// MI455X (gfx1250) — compile-verified
//
#include <hip/hip_runtime.h>

typedef float v2f __attribute__((ext_vector_type(2)));
typedef float v8f __attribute__((ext_vector_type(8)));

// tanh(x) = 1 - 2/(exp(2x)+1); exp via v_exp_f32, rcp via v_rcp_f32.
// Saturates correctly: x->+inf => 1, x->-inf => -1.
__device__ __forceinline__ float tanh_fast(float x) {
    float e = __expf(2.0f * x);
    return 1.0f - 2.0f * __builtin_amdgcn_rcpf(e + 1.0f);
}

// Accumulate y[16 rows] += W[rowtop:rowtop+16, 0:4*NK] * h[0:4*NK] into the
// 16x16 f32 WMMA accumulator (all 16 columns of B hold the same h chunk, so
// all 16 columns of D are identical and equal the matvec result).
// A layout (f32 16x4): lanes 0-15 -> M=lane, VGPR0/1 = K0,K1
//                      lanes 16-31 -> M=lane-16, VGPR0/1 = K2,K3
// B layout (f32 4x16): lanes 0-15 -> col=lane, VGPR0/1 = K0,K1 rows
//                      lanes 16-31 -> col=lane-16, VGPR0/1 = K2,K3 rows
template<int NK>
__device__ __forceinline__ v8f wmma_mv(const float* W, int ldw, int rowtop,
                                       const float* h, int l15, int hi2, v8f c) {
    const float* wrow = W + (rowtop + l15) * ldw + hi2;
    const float* hh   = h + hi2;
#pragma unroll
    for (int k = 0; k < NK; ++k) {
        v2f a = *(const v2f*)(wrow + 4 * k);
        v2f b = *(const v2f*)(hh + 4 * k);
        c = __builtin_amdgcn_wmma_f32_16x16x4_f32(false, a, false, b,
                                                  (short)0, c, false, false);
    }
    return c;
}

// ---- LDS layout (float offsets); all even => 8-byte aligned for b64 loads
#define O_W1HH0  0        // 4096  rnn1 layer0 Whh (64x64)
#define O_W1IH1  4096     // 4096  rnn1 layer1 Wih
#define O_W1HH1  8192     // 4096  rnn1 layer1 Whh
#define O_W2HH0  12288    // 4096  rnn2 layer0 Whh
#define O_W2IH1  16384    // 4096  rnn2 layer1 Wih
#define O_W2HH1  20480    // 4096  rnn2 layer1 Whh
#define O_W2IH0  24576    // 1280  rnn2 layer0 Wih (64x20)
#define O_XBUF   25856    // 200   x
#define O_H1BUF  26056    // 200   lin1 outputs (rnn2 input), 10x20
#define O_B0S1   26256    // 64    bih0+bhh0 (rnn1)
#define O_B1S1   26320    // 64    bih1+bhh1 (rnn1)
#define O_B0S2   26384    // 64    bih0+bhh0 (rnn2)
#define O_B1S2   26448    // 64    bih1+bhh1 (rnn2)
#define O_WIH0C  26512    // 64    rnn1 layer0 Wih column (64x1)
#define O_LIN1W  26576    // 64
#define O_LIN3W  26640    // 320
#define O_H10    26960    // 64    rnn1 layer0 hidden
#define O_H11    27024    // 64    rnn1 layer1 hidden
#define O_H20    27088    // 64    rnn2 layer0 hidden
#define O_H21    27152    // 64    rnn2 layer1 hidden
#define SMEM_FLOATS 27216

__global__ void __launch_bounds__(128)
rnn_persistent_wmma(const float* __restrict__ x,
                    const float* __restrict__ hp1,
                    const float* __restrict__ hp2,
                    const float* __restrict__ w1ih0, const float* __restrict__ w1hh0,
                    const float* __restrict__ b1ih0, const float* __restrict__ b1hh0,
                    const float* __restrict__ w1ih1, const float* __restrict__ w1hh1,
                    const float* __restrict__ b1ih1, const float* __restrict__ b1hh1,
                    const float* __restrict__ w2ih0, const float* __restrict__ w2hh0,
                    const float* __restrict__ b2ih0, const float* __restrict__ b2hh0,
                    const float* __restrict__ w2ih1, const float* __restrict__ w2hh1,
                    const float* __restrict__ b2ih1, const float* __restrict__ b2hh1,
                    const float* __restrict__ lin1W, const float* __restrict__ lin1b,
                    const float* __restrict__ lin3W, const float* __restrict__ lin3b,
                    float* __restrict__ out)
{
    extern __shared__ float sm[];
    const int tid  = threadIdx.x;
    const int w    = tid >> 5;        // wave id 0..3 -> output rows [16w,16w+16)
    const int lane = tid & 31;
    const int l15  = lane & 15;
    const int hi2  = (lane & 16) ? 2 : 0;          // float offset for K2,K3 half
    const int rowsel = 16 * w + ((lane & 16) ? 8 : 0); // C/D: M = r (+8 for hi lanes)

    // ---- stage everything into LDS
    for (int i = tid; i < 4096; i += 128) {
        sm[O_W1HH0 + i] = w1hh0[i];
        sm[O_W1IH1 + i] = w1ih1[i];
        sm[O_W1HH1 + i] = w1hh1[i];
        sm[O_W2HH0 + i] = w2hh0[i];
        sm[O_W2IH1 + i] = w2ih1[i];
        sm[O_W2HH1 + i] = w2hh1[i];
    }
    for (int i = tid; i < 1280; i += 128) sm[O_W2IH0 + i] = w2ih0[i];
    for (int i = tid; i < 200;  i += 128) sm[O_XBUF  + i] = x[i];
    for (int i = tid; i < 320;  i += 128) sm[O_LIN3W + i] = lin3W[i];
    if (tid < 64) {
        sm[O_B0S1 + tid]  = b1ih0[tid] + b1hh0[tid];
        sm[O_B1S1 + tid]  = b1ih1[tid] + b1hh1[tid];
        sm[O_B0S2 + tid]  = b2ih0[tid] + b2hh0[tid];
        sm[O_B1S2 + tid]  = b2ih1[tid] + b2hh1[tid];
        sm[O_WIH0C + tid] = w1ih0[tid];
        sm[O_LIN1W + tid] = lin1W[tid];
        sm[O_H10 + tid] = hp1[tid];
        sm[O_H11 + tid] = hp1[64 + tid];
        sm[O_H20 + tid] = hp2[tid];
        sm[O_H21 + tid] = hp2[64 + tid];
    }
    const float l1b = lin1b[0];
    __syncthreads();

    // ================= rnn1: 10 outer rows x 20 timesteps =================
    for (int tau = 0; tau < 10; ++tau) {
        for (int t = 0; t < 20; ++t) {
            const float xt = sm[O_XBUF + tau * 20 + t];

            // ---- layer 0: z = Wih0*x_t + b + Whh0*h10
            v8f c;
#pragma unroll
            for (int r = 0; r < 8; ++r)
                c[r] = sm[O_B0S1 + rowsel + r] + sm[O_WIH0C + rowsel + r] * xt;
            c = wmma_mv<16>(sm + O_W1HH0, 64, 16 * w, sm + O_H10, l15, hi2, c);
            float y[8];
#pragma unroll
            for (int r = 0; r < 8; ++r) y[r] = tanh_fast(c[r]);
            __syncthreads();                     // everyone done reading h10
            if (l15 == 0) {
#pragma unroll
                for (int r = 0; r < 8; ++r) sm[O_H10 + rowsel + r] = y[r];
            }
            __syncthreads();                     // new h10 visible

            // ---- layer 1: z = Wih1*h10_new + b + Whh1*h11
#pragma unroll
            for (int r = 0; r < 8; ++r) c[r] = sm[O_B1S1 + rowsel + r];
            c = wmma_mv<16>(sm + O_W1IH1, 64, 16 * w, sm + O_H10, l15, hi2, c);
            c = wmma_mv<16>(sm + O_W1HH1, 64, 16 * w, sm + O_H11, l15, hi2, c);
#pragma unroll
            for (int r = 0; r < 8; ++r) y[r] = tanh_fast(c[r]);
            __syncthreads();                     // everyone done reading h11
            if (l15 == 0) {
#pragma unroll
                for (int r = 0; r < 8; ++r) sm[O_H11 + rowsel + r] = y[r];
            }
            __syncthreads();                     // new h11 visible

            // ---- lin1: scalar output -> h1buf (wave0 shuffle reduction)
            if (w == 0) {
                float p = sm[O_LIN1W + lane]      * sm[O_H11 + lane]
                        + sm[O_LIN1W + lane + 32] * sm[O_H11 + lane + 32];
#pragma unroll
                for (int m = 16; m >= 1; m >>= 1) p += __shfl_xor(p, m, 32);
                if (lane == 0) sm[O_H1BUF + tau * 20 + t] = p + l1b;
            }
        }
    }
    __syncthreads();
    if (tid < 64) {                               // hp1_final
        out[50 + tid]  = sm[O_H10 + tid];
        out[114 + tid] = sm[O_H11 + tid];
    }

    // ================= rnn2: 10 timesteps over h1buf rows =================
    for (int tau = 0; tau < 10; ++tau) {
        const float* u = sm + O_H1BUF + tau * 20;   // 20-wide input vector

        // ---- layer 0: z = Wih0(64x20)*u + b + Whh0*h20  (20 = 5 K-chunks of 4)
        v8f c;
#pragma unroll
        for (int r = 0; r < 8; ++r) c[r] = sm[O_B0S2 + rowsel + r];
        c = wmma_mv<5>(sm + O_W2IH0, 20, 16 * w, u, l15, hi2, c);
        c = wmma_mv<16>(sm + O_W2HH0, 64, 16 * w, sm + O_H20, l15, hi2, c);
        float y[8];
#pragma unroll
        for (int r = 0; r < 8; ++r) y[r] = tanh_fast(c[r]);
        __syncthreads();
        if (l15 == 0) {
#pragma unroll
            for (int r = 0; r < 8; ++r) sm[O_H20 + rowsel + r] = y[r];
        }
        __syncthreads();

        // ---- layer 1
#pragma unroll
        for (int r = 0; r < 8; ++r) c[r] = sm[O_B1S2 + rowsel + r];
        c = wmma_mv<16>(sm + O_W2IH1, 64, 16 * w, sm + O_H20, l15, hi2, c);
        c = wmma_mv<16>(sm + O_W2HH1, 64, 16 * w, sm + O_H21, l15, hi2, c);
#pragma unroll
        for (int r = 0; r < 8; ++r) y[r] = tanh_fast(c[r]);
        __syncthreads();
        if (l15 == 0) {
#pragma unroll
            for (int r = 0; r < 8; ++r) sm[O_H21 + rowsel + r] = y[r];
        }
        __syncthreads();

        // ---- lin3: 5 outputs for this step (wave0)
        if (w == 0) {
#pragma unroll
            for (int o = 0; o < 5; ++o) {
                float p = sm[O_LIN3W + o * 64 + lane]      * sm[O_H21 + lane]
                        + sm[O_LIN3W + o * 64 + lane + 32] * sm[O_H21 + lane + 32];
#pragma unroll
                for (int m = 16; m >= 1; m >>= 1) p += __shfl_xor(p, m, 32);
                if (lane == 0) out[tau * 5 + o] = p + lin3b[o];
            }
        }
    }
    __syncthreads();
    if (tid < 64) {                               // hp2_final
        out[178 + tid] = sm[O_H20 + tid];
        out[242 + tid] = sm[O_H21 + tid];
    }
}

extern "C" void kernel_launch(void* const* d_in, const int* in_sizes, int n_in,
                              void* d_out, int out_size, void* d_ws, size_t ws_size,
                              hipStream_t stream) {
    (void)in_sizes; (void)n_in; (void)d_ws; (void)ws_size; (void)out_size;
    const float* X     = (const float*)d_in[0];
    const float* HP1   = (const float*)d_in[1];
    const float* HP2   = (const float*)d_in[2];
    const float* W1IH0 = (const float*)d_in[3];
    const float* W1HH0 = (const float*)d_in[4];
    const float* B1IH0 = (const float*)d_in[5];
    const float* B1HH0 = (const float*)d_in[6];
    const float* W1IH1 = (const float*)d_in[7];
    const float* W1HH1 = (const float*)d_in[8];
    const float* B1IH1 = (const float*)d_in[9];
    const float* B1HH1 = (const float*)d_in[10];
    const float* W2IH0 = (const float*)d_in[11];
    const float* W2HH0 = (const float*)d_in[12];
    const float* B2IH0 = (const float*)d_in[13];
    const float* B2HH0 = (const float*)d_in[14];
    const float* W2IH1 = (const float*)d_in[15];
    const float* W2HH1 = (const float*)d_in[16];
    const float* B2IH1 = (const float*)d_in[17];
    const float* B2HH1 = (const float*)d_in[18];
    const float* L1W   = (const float*)d_in[19];
    const float* L1B   = (const float*)d_in[20];
    const float* L3W   = (const float*)d_in[21];
    const float* L3B   = (const float*)d_in[22];
    float* OUT = (float*)d_out;

    const size_t smem = SMEM_FLOATS * sizeof(float);  // ~106 KB < 320 KB/WGP
    rnn_persistent_wmma<<<1, 128, smem, stream>>>(
        X, HP1, HP2,
        W1IH0, W1HH0, B1IH0, B1HH0, W1IH1, W1HH1, B1IH1, B1HH1,
        W2IH0, W2HH0, B2IH0, B2HH0, W2IH1, W2HH1, B2IH1, B2HH1,
        L1W, L1B, L3W, L3B, OUT);
}